// Dtopk_22299470201291
// MI455X (gfx1250) — compile-verified
//
#include <hip/hip_runtime.h>

// ---------------------------------------------------------------------------
// Problem constants (from reference: x is (128, 16384) fp32, num == 32)
// ---------------------------------------------------------------------------
#define TPB1 1024          // threads per block, kernel 1 (32 waves)
#define NW1  (TPB1 / 32)
#define TPB2 256           // threads per block, kernel 2 (8 waves)
#define NW2  (TPB2 / 32)
#define NCHUNK 8           // 32-column groups per wave in kernel 2

constexpr int   NCOL   = 16384;
constexpr int   KTOP   = 32;
// float32 eps = 2^-23  ->  eps^2 = 2^-46 (exactly representable)
constexpr float E2     = 1.4210854715202004e-14f;  // 2^-46
constexpr float INV_E2 = 7.0368744177664e13f;      // 2^46 (exact pow2 scale)

typedef int   v4i __attribute__((ext_vector_type(4)));
typedef float v4f __attribute__((ext_vector_type(4)));
typedef __attribute__((address_space(1))) v4i glob_v4i;
typedef __attribute__((address_space(3))) v4i lds_v4i;

// ---------------------------------------------------------------------------
// CDNA5 async global->LDS copy (ASYNCcnt-tracked) with fallbacks
// ---------------------------------------------------------------------------
__device__ __forceinline__ void async_copy_b128(void* lds, const void* gsrc) {
#if defined(__gfx1250__) && __has_builtin(__builtin_amdgcn_global_load_async_to_lds_b128)
  __builtin_amdgcn_global_load_async_to_lds_b128(
      (glob_v4i*)(void*)gsrc,
      (lds_v4i*)lds,
      0, 0);
#elif defined(__gfx1250__)
  unsigned lds_off = (unsigned)(unsigned long long)lds;  // addr[31:0] == LDS offset
  asm volatile("global_load_async_to_lds_b128 %0, %1, off"
               :: "v"(lds_off), "v"(gsrc) : "memory");
#else
  *(v4f*)lds = *(const v4f*)gsrc;
#endif
}

__device__ __forceinline__ void wait_async0() {
#if defined(__gfx1250__) && __has_builtin(__builtin_amdgcn_s_wait_asynccnt)
  __builtin_amdgcn_s_wait_asynccnt(0);
#elif defined(__gfx1250__)
  asm volatile("s_wait_asynccnt 0" ::: "memory");
#endif
}

// out = relu( relu(e2 - d^2)/e2 - s + 1 ),  d = x - t   (matches ref op order)
__device__ __forceinline__ float soft_ind(float xb, float t, float s) {
  float d   = xb - t;
  float dd  = d * d;
  float q   = E2 - dd;
  float ind = fmaxf(q, 0.0f) * INV_E2;
  return fmaxf((ind - s) + 1.0f, 0.0f);
}

// ---------------------------------------------------------------------------
// Kernel 1: one block per row. Async-stage row in LDS, iterative top-32
// extraction (argmax w/ lowest-index tie-break, mask one occurrence each
// round), then per-wave deterministic sum  s_j = sum_n relu(e2-d^2)/e2.
// Writes [t[0..31], s[0..31]] per row into workspace.
// ---------------------------------------------------------------------------
__global__ __launch_bounds__(TPB1)
void topk_stats_kernel(const float* __restrict__ x, float* __restrict__ ts) {
  __shared__ __align__(16) float row[NCOL];   // 64 KB (LDS/WGP = 320 KB)
  __shared__ float redv[NW1];
  __shared__ int   redi[NW1];
  __shared__ float tvals[KTOP];
  __shared__ int   tidxs[KTOP];

  const int b    = blockIdx.x;
  const int tid  = threadIdx.x;
  const int lane = tid & 31;
  const int wave = tid >> 5;
  const float* xr = x + (size_t)b * NCOL;

  // Stage the row into LDS via async B128 copies (4 per thread).
  for (int i = tid * 4; i < NCOL; i += TPB1 * 4)
    async_copy_b128(&row[i], &xr[i]);
  wait_async0();
  __syncthreads();

  // 32 rounds of argmax extraction.
  for (int it = 0; it < KTOP; ++it) {
    float bv = -__builtin_inff();
    int   bi = 0x7fffffff;
    for (int i = tid; i < NCOL; i += TPB1) {
      float v = row[i];
      if (v > bv) { bv = v; bi = i; }
    }
    for (int off = 16; off > 0; off >>= 1) {
      float ov = __shfl_down(bv, off, 32);
      int   oi = __shfl_down(bi, off, 32);
      if (ov > bv || (ov == bv && oi < bi)) { bv = ov; bi = oi; }
    }
    if (lane == 0) { redv[wave] = bv; redi[wave] = bi; }
    __syncthreads();
    if (wave == 0) {
      bv = redv[lane]; bi = redi[lane];
      for (int off = 16; off > 0; off >>= 1) {
        float ov = __shfl_down(bv, off, 32);
        int   oi = __shfl_down(bi, off, 32);
        if (ov > bv || (ov == bv && oi < bi)) { bv = ov; bi = oi; }
      }
      if (lane == 0) {
        tvals[it] = bv;
        tidxs[it] = bi;
        row[bi]   = -__builtin_inff();   // remove exactly one occurrence
      }
    }
    __syncthreads();
  }

  // Restore the 32 masked entries so the sum sees the original row.
  if (tid < KTOP) row[tidxs[tid]] = tvals[tid];
  __syncthreads();

  // Wave j computes s_j (deterministic lane-strided + shuffle reduction).
  {
    const float tj = tvals[wave];
    float acc = 0.0f;
    for (int i = lane; i < NCOL; i += 32) {
      float d  = row[i] - tj;
      float dd = d * d;
      float q  = E2 - dd;
      acc += fmaxf(q, 0.0f) * INV_E2;   // exact pow2 scale, matches ref order
    }
    for (int off = 16; off > 0; off >>= 1)
      acc += __shfl_down(acc, off, 32);
    if (lane == 0) redv[wave] = acc;    // reuse redv as s-values
  }
  __syncthreads();

  if (tid < KTOP) {
    float* o = ts + (size_t)b * (2 * KTOP);
    o[tid]        = tvals[tid];
    o[KTOP + tid] = redv[tid];
  }
}

// ---------------------------------------------------------------------------
// Kernel 2: stream the (B, N, 32) output with b128 NT stores.
// Lane layout: kg = lane&7 owns k = 4*kg..4*kg+3 (one aligned float4 of the
// output row); nsub = lane>>3 picks which of 4 n's the lane serves. One wave
// instruction retires 128 output elements; every n's 32 k-values are written
// as 8 lanes x 16B = 128B contiguous.
// ---------------------------------------------------------------------------
__global__ __launch_bounds__(TPB2)
void expand_kernel(const float* __restrict__ x, const float* __restrict__ ts,
                   float* __restrict__ out) {
  const int b    = blockIdx.y;
  const int tid  = threadIdx.x;
  const int lane = tid & 31;
  const int wave = tid >> 5;
  const int kg   = lane & 7;    // k-group: k = 4*kg .. 4*kg+3
  const int nsub = lane >> 3;   // which n within each quad (0..3)

  const float* tsrow = ts + (size_t)b * (2 * KTOP);
  const v4f tl = *(const v4f*)&tsrow[4 * kg];          // t[4kg..4kg+3]
  const v4f sl = *(const v4f*)&tsrow[KTOP + 4 * kg];   // s[4kg..4kg+3]

  const float* xr   = x + (size_t)b * NCOL;
  float*       orow = out + (size_t)b * NCOL * KTOP;

  const int n0 = (blockIdx.x * NW2 + wave) * (32 * NCHUNK);
  for (int c = 0; c < NCHUNK; ++c) {
    const int nb = n0 + c * 32;
    const float xv = xr[nb + lane];          // coalesced: 32 consecutive n
#pragma unroll
    for (int m = 0; m < 8; ++m) {
      const int n  = nb + 4 * m + nsub;      // 4 n's per wave-iteration
      const float xb = __shfl(xv, 4 * m + nsub, 32);
      v4f o;
      o.x = soft_ind(xb, tl.x, sl.x);
      o.y = soft_ind(xb, tl.y, sl.y);
      o.z = soft_ind(xb, tl.z, sl.z);
      o.w = soft_ind(xb, tl.w, sl.w);
      // 16B-aligned v4f NT store: out[b][n][4kg .. 4kg+3]
      __builtin_nontemporal_store(o, (v4f*)&orow[(size_t)n * KTOP + 4 * kg]);
    }
  }
}

// ---------------------------------------------------------------------------
// Entry point
// ---------------------------------------------------------------------------
extern "C" void kernel_launch(void* const* d_in, const int* in_sizes, int n_in,
                              void* d_out, int out_size, void* d_ws, size_t ws_size,
                              hipStream_t stream) {
  const float* x  = (const float*)d_in[0];   // (B, 16384) fp32
  // d_in[1] is the python scalar num == 32 (compile-time constant here)
  float* out = (float*)d_out;                // (B, 16384, 32) fp32
  float* ts  = (float*)d_ws;                 // per-row [t[32], s[32]]

  const int B = in_sizes[0] / NCOL;

  topk_stats_kernel<<<B, TPB1, 0, stream>>>(x, ts);

  const int gx = NCOL / (NW2 * 32 * NCHUNK);   // 16384 / 2048 = 8
  expand_kernel<<<dim3(gx, B), TPB2, 0, stream>>>(x, ts, out);
}